// MultimodalGNN_10874857193675
// MI455X (gfx1250) — compile-verified
//
#include <hip/hip_runtime.h>
#include <hip/hip_bf16.h>

// ---------------- problem constants ----------------
#define BSZ 4
#define FRAMES 8
#define CH 128
#define FF 196          // H*W = 14*14
#define HEADS 4
#define NUM_REL 7
#define NNODE 72        // bs * (2*frames + 2)
#define ETOT 453        // 4*96 structural + 69 self loops
#define TEXTLEN 300

typedef __attribute__((ext_vector_type(2))) float v2f;
typedef __attribute__((ext_vector_type(8))) float v8f;

// ---------------- graph init (deterministic, single thread) ----------------
__global__ void init_graph(int* __restrict__ gsrc, int* __restrict__ gdst,
                           int* __restrict__ get, float* __restrict__ gnorm,
                           int* __restrict__ csr_off, int* __restrict__ csr_idx) {
    if (threadIdx.x != 0 || blockIdx.x != 0) return;
    int E = 0;
    for (int b = 0; b < BSZ; ++b) {
        int s = b * (2 * FRAMES + 1);       // stride-17 (replicates reference bug)
        int tot = s + 2 * FRAMES + 1;
        int ff = s + FRAMES;
        for (int img = s; img < ff; ++img) {
            int mo = ff + (img - s);
            if (img < ff - 1) {
                gsrc[E] = img; gdst[E] = img + 1; get[E] = 0; ++E;
                gsrc[E] = mo;  gdst[E] = mo + 1;  get[E] = 1; ++E;
            }
            gsrc[E] = img;    gdst[E] = mo;      get[E] = 2; ++E;
            gsrc[E] = mo;     gdst[E] = img;     get[E] = 2; ++E;
            gsrc[E] = img;    gdst[E] = tot - 1; get[E] = 3; ++E;
            gsrc[E] = tot - 1;gdst[E] = img;     get[E] = 3; ++E;
            gsrc[E] = mo;     gdst[E] = tot - 1; get[E] = 4; ++E;
            gsrc[E] = tot - 1;gdst[E] = mo;      get[E] = 4; ++E;
        }
        int g = tot;
        for (int node = s; node < tot; ++node) {
            gsrc[E] = g;    gdst[E] = node; get[E] = 5; ++E;
            gsrc[E] = node; gdst[E] = g;    get[E] = 5; ++E;
        }
    }
    for (int nn = 0; nn < 69; ++nn) {       // self loops nodes 0..68, type 6
        gsrc[E] = nn; gdst[E] = nn; get[E] = 6; ++E;
    }
    int cnt[NNODE * NUM_REL];
    for (int i = 0; i < NNODE * NUM_REL; ++i) cnt[i] = 0;
    for (int e = 0; e < ETOT; ++e) cnt[gdst[e] * NUM_REL + get[e]]++;
    for (int e = 0; e < ETOT; ++e) {
        int c = cnt[gdst[e] * NUM_REL + get[e]];
        gnorm[e] = 1.0f / (float)(c < 1 ? 1 : c);
    }
    int deg[NNODE];
    for (int i = 0; i < NNODE; ++i) deg[i] = 0;
    for (int e = 0; e < ETOT; ++e) deg[gdst[e]]++;
    csr_off[0] = 0;
    for (int i = 0; i < NNODE; ++i) csr_off[i + 1] = csr_off[i] + deg[i];
    int cur[NNODE];
    for (int i = 0; i < NNODE; ++i) cur[i] = csr_off[i];
    for (int e = 0; e < ETOT; ++e) csr_idx[cur[gdst[e]]++] = e;
}

// ---------------- node feature construction: x0[72,128,196] ----------------
__global__ void build_nodes(const float* __restrict__ vis, const float* __restrict__ mot,
                            const float* __restrict__ txt, float* __restrict__ x0) {
    int idx = blockIdx.x * blockDim.x + threadIdx.x;  // c*FF + f
    if (idx >= CH * FF) return;
    int b = blockIdx.z, j = blockIdx.y;               // j in 0..17
    int c = idx / FF, f = idx % FF;
    float val;
    if (j < FRAMES) {
        val = vis[(((long)b * FRAMES + j) * CH + c) * FF + f];
    } else if (j < 2 * FRAMES) {
        val = mot[(((long)b * FRAMES + (j - FRAMES)) * CH + c) * FF + f];
    } else if (j == 2 * FRAMES) {
        val = txt[((long)b * CH + c) * TEXTLEN + f];  // tgt<300 -> plain slice
    } else {
        float sum = 0.f;
        for (int q = 0; q < FRAMES; ++q)
            sum += vis[(((long)b * FRAMES + q) * CH + c) * FF + f]
                 + mot[(((long)b * FRAMES + q) * CH + c) * FF + f];
        sum += txt[((long)b * CH + c) * TEXTLEN + f];
        val = sum / 17.0f;
    }
    x0[(((long)b * 18 + j) * CH + c) * FF + f] = 2.0f * val;  // feat+pos => 2x
}

// ---------------- head mean: xr[n,o,f] = mean_h xl[n,h,o,f] ----------------
__global__ void head_mean(const float* __restrict__ xl, float* __restrict__ xr, int OD) {
    long idx = (long)blockIdx.x * blockDim.x + threadIdx.x;  // over NNODE*OD*FF
    long tot = (long)NNODE * OD * FF;
    if (idx >= tot) return;
    long per = (long)OD * FF;
    int n = (int)(idx / per);
    long rem = idx % per;
    const float* p = xl + (long)n * HEADS * per + rem;
    xr[idx] = 0.25f * (p[0] + p[per] + p[2 * per] + p[3 * per]);
}

// ---------------- WMMA f32 GEMM: C[n][M,F] (+)= A x B[n] ----------------
// TRANSA=0: A row-major [M,KD]. TRANSA=1: A[k*M + m] (i.e. stored [KD,M]).
// Each single-wave block computes MT vertically-stacked 16x16 tiles (4 indep WMMA chains).
template <int KD, int MT, int TRANSA>
__global__ void gemm_wmma(const float* __restrict__ A, const float* __restrict__ Bbase,
                          float* __restrict__ Cbase, int M,
                          long bStride, long cStride,
                          const float* __restrict__ bias, int accumulate) {
    int ft = blockIdx.x, mt = blockIdx.y, n = blockIdx.z;
    const float* B = Bbase + (long)n * bStride;
    float* C = Cbase + (long)n * cStride;
    int l = threadIdx.x;          // 0..31
    int hi = l >> 4, lo = l & 15;
    int f0 = ft * 16;
    int m0 = mt * (MT * 16);
    int fc = f0 + lo; if (fc >= FF) fc = FF - 1;   // clamp: garbage cols masked at store
    const float* bp = B + (long)(2 * hi) * FF + fc;  // rows k = 2*hi + kk (+v)
    v8f acc[MT] = {};
    for (int kk = 0; kk < KD; kk += 4) {
        v2f b;
        b[0] = bp[(long)kk * FF];
        b[1] = bp[(long)(kk + 1) * FF];
#pragma unroll
        for (int t = 0; t < MT; ++t) {
            v2f a;
            if (TRANSA) {
                int m = m0 + t * 16 + lo;
                a[0] = A[(long)(kk + 2 * hi) * M + m];
                a[1] = A[(long)(kk + 1 + 2 * hi) * M + m];
            } else {
                a = *(const v2f*)(A + (long)(m0 + t * 16 + lo) * KD + kk + 2 * hi);
            }
            acc[t] = __builtin_amdgcn_wmma_f32_16x16x4_f32(false, a, false, b, (short)0,
                                                           acc[t], false, false);
        }
    }
    int f = f0 + lo;
    if (f < FF) {
#pragma unroll
        for (int t = 0; t < MT; ++t)
#pragma unroll
            for (int v = 0; v < 8; ++v) {
                int m = m0 + t * 16 + v + 8 * hi;
                float r = acc[t][v];
                if (bias) r += bias[m];
                long idx = (long)m * FF + f;
                if (accumulate) C[idx] += r; else C[idx] = r;
            }
    }
}

// ---------------- attention logits: alpha[e,h,f] ----------------
__global__ void attn_alpha(const float* __restrict__ xl, const float* __restrict__ att,
                           const int* __restrict__ gsrc, const int* __restrict__ gdst,
                           float* __restrict__ alpha, int OD) {
    int e = blockIdx.x, h = blockIdx.y, f = threadIdx.x;
    if (f >= FF) return;
    const float* xs = xl + ((long)gsrc[e] * HEADS + h) * OD * FF + f;
    const float* xd = xl + ((long)gdst[e] * HEADS + h) * OD * FF + f;
    const float* aw = att + h * OD;
    float acc = 0.f;
    for (int o = 0; o < OD; ++o) {
        float v = xs[(long)o * FF] + xd[(long)o * FF];
        float lr = v > 0.f ? v : 0.2f * v;       // leaky_relu(0.2)
        acc += lr * aw[o];
    }
    alpha[((long)e * HEADS + h) * FF + f] = acc;
}

// ---------------- softmax over incoming edges per (dst,h,f) ----------------
__global__ void attn_softmax(float* __restrict__ alpha, const int* __restrict__ csr_off,
                             const int* __restrict__ csr_idx) {
    int n = blockIdx.x, h = blockIdx.y, f = threadIdx.x;
    if (f >= FF) return;
    int b0 = csr_off[n], b1 = csr_off[n + 1];
    float m = -3.4e38f;
    for (int j = b0; j < b1; ++j)
        m = fmaxf(m, alpha[((long)csr_idx[j] * HEADS + h) * FF + f]);
    float den = 0.f;
    for (int j = b0; j < b1; ++j) {
        long idx = ((long)csr_idx[j] * HEADS + h) * FF + f;
        float ex = __expf(alpha[idx] - m);
        alpha[idx] = ex; den += ex;
    }
    float inv = 1.0f / (den + 1e-16f);
    for (int j = b0; j < b1; ++j)
        alpha[((long)csr_idx[j] * HEADS + h) * FF + f] *= inv;
}

// ---------------- per-edge relation GEMM + scatter-add ----------------
// D_e[p,f] = sum_o rel_w[et][o,p] * meanh( xl[src,h,o,f]*alpha[e,h,f] )
// out[dst] += norm_e * D_e
template <int OD, int PT>  // PT = OD/16
__global__ void edge_msg_wmma(const float* __restrict__ xl, const float* __restrict__ alpha,
                              const float* __restrict__ rel_w,
                              const int* __restrict__ gsrc, const int* __restrict__ gdst,
                              const int* __restrict__ get, const float* __restrict__ gnorm,
                              float* __restrict__ out) {
    __shared__ float Bs[OD * 16];
    int e = blockIdx.x, ft = blockIdx.y;
    int l = threadIdx.x;
    int hi = l >> 4, lo = l & 15;
    int f0 = ft * 16;
    int s = gsrc[e], d = gdst[e], r = get[e];
    float nrm = gnorm[e];
    int fc = f0 + lo; if (fc >= FF) fc = FF - 1;
    float a0 = alpha[((long)e * HEADS + 0) * FF + fc];
    float a1 = alpha[((long)e * HEADS + 1) * FF + fc];
    float a2 = alpha[((long)e * HEADS + 2) * FF + fc];
    float a3 = alpha[((long)e * HEADS + 3) * FF + fc];
    const float* xs = xl + (long)s * HEADS * OD * FF + fc;
#pragma unroll 4
    for (int o = hi; o < OD; o += 2) {
        float v = xs[(long)(0 * OD + o) * FF] * a0
                + xs[(long)(1 * OD + o) * FF] * a1
                + xs[(long)(2 * OD + o) * FF] * a2
                + xs[(long)(3 * OD + o) * FF] * a3;
        Bs[o * 16 + lo] = 0.25f * v;   // head mean
    }
    __syncthreads();
    const float* W = rel_w + (long)r * OD * OD + lo;  // W[o,p]; A[m=p,k=o]=W[o*OD+p]
    v8f acc[PT] = {};
    for (int kk = 0; kk < OD; kk += 4) {
        v2f b;
        b[0] = Bs[(kk + 0 + 2 * hi) * 16 + lo];
        b[1] = Bs[(kk + 1 + 2 * hi) * 16 + lo];
#pragma unroll
        for (int p = 0; p < PT; ++p) {
            v2f a;
            a[0] = W[(long)(kk + 0 + 2 * hi) * OD + p * 16];
            a[1] = W[(long)(kk + 1 + 2 * hi) * OD + p * 16];
            acc[p] = __builtin_amdgcn_wmma_f32_16x16x4_f32(false, a, false, b, (short)0,
                                                           acc[p], false, false);
        }
    }
    int f = f0 + lo;
    if (f < FF) {
        float* outn = out + (long)d * OD * FF;
#pragma unroll
        for (int p = 0; p < PT; ++p)
#pragma unroll
            for (int v = 0; v < 8; ++v) {
                int m = p * 16 + v + 8 * hi;
                atomicAdd(&outn[(long)m * FF + f], nrm * acc[p][v]);
            }
    }
}

// ---------------- readout ----------------
__global__ void readout(const float* __restrict__ vis, const float* __restrict__ x2,
                        float* __restrict__ out) {
    long idx = (long)blockIdx.x * blockDim.x + threadIdx.x;
    const long featN = (long)BSZ * FRAMES * CH * FF;   // 802816
    const long totN = featN + (long)BSZ * CH * FF;     // 903168
    if (idx >= totN) return;
    if (idx < featN) {
        long bf = idx / (CH * FF);
        long rem = idx % (CH * FF);
        int b = (int)(bf / FRAMES), fr = (int)(bf % FRAMES);
        int node = b * (2 * FRAMES + 1) + fr;          // stride-17 readout
        out[idx] = vis[idx] + x2[(long)node * CH * FF + rem];
    } else {
        long k = idx - featN;
        int b = (int)(k / (CH * FF));
        long rem = k % (CH * FF);
        int node = b * (2 * FRAMES + 1) + 2 * FRAMES + 2;  // = b*17 + 18
        out[idx] = x2[(long)node * CH * FF + rem];
    }
}

// ---------------- launcher ----------------
extern "C" void kernel_launch(void* const* d_in, const int* in_sizes, int n_in,
                              void* d_out, int out_size, void* d_ws, size_t ws_size,
                              hipStream_t stream) {
    (void)in_sizes; (void)n_in; (void)out_size; (void)ws_size;
    const float* vis   = (const float*)d_in[0];
    const float* mot   = (const float*)d_in[2];
    const float* txt   = (const float*)d_in[4];
    const float* lin_w1  = (const float*)d_in[6];
    const float* att1    = (const float*)d_in[7];
    const float* rel_w1  = (const float*)d_in[8];
    const float* root_w1 = (const float*)d_in[9];
    const float* bias1   = (const float*)d_in[10];
    const float* lin_w2  = (const float*)d_in[11];
    const float* att2    = (const float*)d_in[12];
    const float* rel_w2  = (const float*)d_in[13];
    const float* root_w2 = (const float*)d_in[14];
    const float* bias2   = (const float*)d_in[15];
    float* out = (float*)d_out;

    char* wsb = (char*)d_ws;
    int*   g_src  = (int*)(wsb + 0);
    int*   g_dst  = (int*)(wsb + 2048);
    int*   g_et   = (int*)(wsb + 4096);
    float* g_norm = (float*)(wsb + 6144);
    int*   g_coff = (int*)(wsb + 8192);
    int*   g_cidx = (int*)(wsb + 9216);
    float* x0    = (float*)(wsb + ((size_t)1 << 16));   //  7.23 MB
    float* xl    = (float*)(wsb + ((size_t)8 << 20));   // 57.8 MB (max of both layers)
    float* alpha = (float*)(wsb + ((size_t)66 << 20));  //  1.42 MB
    float* x1    = (float*)(wsb + ((size_t)68 << 20));  // 14.45 MB (layer1 output)
    float* x2    = (float*)(wsb + ((size_t)83 << 20));  //  7.23 MB (layer2 output)
    float* xr    = (float*)(wsb + ((size_t)91 << 20));  // 14.45 MB (head mean)

    init_graph<<<1, 1, 0, stream>>>(g_src, g_dst, g_et, g_norm, g_coff, g_cidx);
    build_nodes<<<dim3((CH * FF) / 256, 18, BSZ), 256, 0, stream>>>(vis, mot, txt, x0);

    const int FT = (FF + 15) / 16;   // 13

    // ---- layer 1: Cin=128, o=256, K=1024 ----
    gemm_wmma<128, 4, 0><<<dim3(FT, 1024 / 64, NNODE), 32, 0, stream>>>(
        lin_w1, x0, xl, 1024, (long)CH * FF, (long)1024 * FF, nullptr, 0);
    attn_alpha<<<dim3(ETOT, HEADS), 224, 0, stream>>>(xl, att1, g_src, g_dst, alpha, 256);
    attn_softmax<<<dim3(NNODE, HEADS), 224, 0, stream>>>(alpha, g_coff, g_cidx);
    hipMemsetAsync(x1, 0, (size_t)NNODE * 256 * FF * 4, stream);
    edge_msg_wmma<256, 16><<<dim3(ETOT, FT), 32, 0, stream>>>(
        xl, alpha, rel_w1, g_src, g_dst, g_et, g_norm, x1);
    head_mean<<<(NNODE * 256 * FF) / 256, 256, 0, stream>>>(xl, xr, 256);
    gemm_wmma<256, 4, 1><<<dim3(FT, 256 / 64, NNODE), 32, 0, stream>>>(
        root_w1, xr, x1, 256, (long)256 * FF, (long)256 * FF, bias1, 1);

    // ---- layer 2: Cin=256, o=128, K=512 ----
    gemm_wmma<256, 4, 0><<<dim3(FT, 512 / 64, NNODE), 32, 0, stream>>>(
        lin_w2, x1, xl, 512, (long)256 * FF, (long)512 * FF, nullptr, 0);
    attn_alpha<<<dim3(ETOT, HEADS), 224, 0, stream>>>(xl, att2, g_src, g_dst, alpha, 128);
    attn_softmax<<<dim3(NNODE, HEADS), 224, 0, stream>>>(alpha, g_coff, g_cidx);
    hipMemsetAsync(x2, 0, (size_t)NNODE * 128 * FF * 4, stream);
    edge_msg_wmma<128, 8><<<dim3(ETOT, FT), 32, 0, stream>>>(
        xl, alpha, rel_w2, g_src, g_dst, g_et, g_norm, x2);
    head_mean<<<(NNODE * 128 * FF) / 256, 256, 0, stream>>>(xl, xr, 128);
    gemm_wmma<128, 4, 1><<<dim3(FT, 128 / 64, NNODE), 32, 0, stream>>>(
        root_w2, xr, x2, 128, (long)128 * FF, (long)128 * FF, bias2, 1);

    // ---- readout ----
    const long totN = (long)(BSZ * FRAMES + BSZ) * CH * FF;   // 903168
    readout<<<(int)(totN / 256), 256, 0, stream>>>(vis, x2, out);
}